// NormalLoss_26628797235306
// MI455X (gfx1250) — compile-verified
//
#include <hip/hip_runtime.h>
#include <math.h>

typedef __attribute__((ext_vector_type(2))) float v2f;
typedef __attribute__((ext_vector_type(8))) float v8f;

#define KNN    10
#define NPTS   4096
#define NTILE  (NPTS / 16)
#define NB     8
#define EPSV   1e-8f
#define WAVES  8            // waves per block; block handles 8 query tiles
#define BLOCKT (WAVES * 32)

// sorted-ascending top-k insertion (fully unrolled -> stays in VGPRs)
__device__ __forceinline__ void topk_insert(float d, int idx,
                                            float (&bd)[KNN], int (&bi)[KNN]) {
    if (d < bd[KNN - 1]) {
        bd[KNN - 1] = d; bi[KNN - 1] = idx;
#pragma unroll
        for (int j = KNN - 1; j > 0; --j) {
            if (bd[j] < bd[j - 1]) {
                float td = bd[j]; bd[j] = bd[j - 1]; bd[j - 1] = td;
                int   ti = bi[j]; bi[j] = bi[j - 1]; bi[j - 1] = ti;
            }
        }
    }
}

// 8 waves per block; whole batch staged once in LDS as (x,y,z,|p|^2) float4.
// Each wave owns one 16-query tile; distance tiles via V_WMMA_F32_16X16X4_F32.
__global__ __launch_bounds__(BLOCKT)
void knn_normals_kernel(const float* __restrict__ pts,   // (B, 3, N)
                        float* __restrict__ normals) {   // (B, N, 3)
    __shared__ float4 spts[NPTS];                        // 64 KB of the 320 KB WGP LDS

    const int tid  = threadIdx.x;
    const int wave = tid >> 5;
    const int lane = tid & 31;
    const int half = lane >> 4;          // 0: D rows 0-7, 1: D rows 8-15
    const int l16  = lane & 15;          // column (query) index within tile
    const int b    = blockIdx.y;

    const float* __restrict__ px = pts + (size_t)b * 3 * NPTS;
    const float* __restrict__ py = px + NPTS;
    const float* __restrict__ pz = px + 2 * NPTS;

    // ---- stage the whole cloud into LDS once, precomputing |p|^2 ----
    for (int i = tid; i < NPTS; i += BLOCKT) {
        const float x = px[i], y = py[i], z = pz[i];
        spts[i] = make_float4(x, y, z, fmaf(x, x, fmaf(y, y, z * z)));
    }
    __syncthreads();

    // ---- loop-invariant B matrix (queries) and C matrix (query norms) ----
    const int    q  = (blockIdx.x * WAVES + wave) * 16 + l16;
    const float4 qv = spts[q];

    // B (4x16): VGPR0 = {K=0 lanes0-15, K=2 lanes16-31}, VGPR1 = {K=1, K=3}
    v2f bm;
    bm.x = half ? (-2.0f * qv.z) : (-2.0f * qv.x);
    bm.y = half ? 1.0f           : (-2.0f * qv.y);

    v8f cm;
#pragma unroll
    for (int i = 0; i < 8; ++i) cm[i] = qv.w;   // C[M][N] = |q_N|^2

    float bd[KNN]; int bi[KNN];
#pragma unroll
    for (int j = 0; j < KNN; ++j) { bd[j] = 3.0e38f; bi[j] = 0; }

    // ---- sweep candidate tiles: software-pipelined ds_load_b64 + WMMA ----
    const float2* __restrict__ sp2 = reinterpret_cast<const float2*>(spts);
    const int fidx = 2 * l16 + half;     // lanes 0-15: (x,y), lanes 16-31: (z,|p|^2)

    float2 av = sp2[fidx];               // prologue: operand for tile 0
    for (int t = 0; t < NTILE; ++t) {
        v2f am; am.x = av.x; am.y = av.y;
        // prefetch next tile's A-operand; LDS latency hides under WMMA + top-k
        av = sp2[(((t + 1) & (NTILE - 1)) << 5) + fidx];

        v8f d = __builtin_amdgcn_wmma_f32_16x16x4_f32(
            /*neg_a=*/false, am, /*neg_b=*/false, bm,
            /*c_mod=*/(short)0, cm, /*reuse_a=*/false, /*reuse_b=*/false);

        // single early-out for the whole tile (common case: nothing to insert)
        const float mtile = fminf(fminf(fminf(d[0], d[1]), fminf(d[2], d[3])),
                                  fminf(fminf(d[4], d[5]), fminf(d[6], d[7])));
        if (mtile < bd[KNN - 1]) {
            const int base = (t << 4) + (half << 3); // D row M = i + 8*half
#pragma unroll
            for (int i = 0; i < 8; ++i)
                topk_insert(d[i], base + i, bd, bi);
        }
    }

    // ---- merge top-10 of the two half-lanes covering the same column ----
#pragma unroll
    for (int j = 0; j < KNN; ++j) {
        float od = __shfl_xor(bd[j], 16, 32);
        int   oi = __shfl_xor(bi[j], 16, 32);
        topk_insert(od, oi, bd, bi);
    }

    // ---- gather neighborhood from LDS, covariance ----
    float gx[KNN], gy[KNN], gz[KNN];
    float mx = 0.f, my = 0.f, mz = 0.f;
#pragma unroll
    for (int j = 0; j < KNN; ++j) {
        const float4 gv = spts[bi[j]];
        gx[j] = gv.x; gy[j] = gv.y; gz[j] = gv.z;
        mx += gv.x; my += gv.y; mz += gv.z;
    }
    const float invk = 1.0f / (float)KNN;
    mx *= invk; my *= invk; mz *= invk;

    float a00 = 0.f, a01 = 0.f, a02 = 0.f, a11 = 0.f, a12 = 0.f, a22 = 0.f;
#pragma unroll
    for (int j = 0; j < KNN; ++j) {
        const float dx = gx[j] - mx, dy = gy[j] - my, dz = gz[j] - mz;
        a00 = fmaf(dx, dx, a00); a01 = fmaf(dx, dy, a01); a02 = fmaf(dx, dz, a02);
        a11 = fmaf(dy, dy, a11); a12 = fmaf(dy, dz, a12); a22 = fmaf(dz, dz, a22);
    }
    a00 *= invk; a01 *= invk; a02 *= invk; a11 *= invk; a12 *= invk; a22 *= invk;

    // ---- smallest eigenvalue of symmetric 3x3 (trigonometric formula) ----
    const float q3  = (a00 + a11 + a22) * (1.0f / 3.0f);
    const float p1  = a01 * a01 + a02 * a02 + a12 * a12;
    const float b00 = a00 - q3, b11 = a11 - q3, b22 = a22 - q3;
    const float p2  = b00 * b00 + b11 * b11 + b22 * b22 + 2.0f * p1;
    const float p   = sqrtf(p2 * (1.0f / 6.0f));
    float lmin;
    if (p > 1e-20f) {
        const float ip  = 1.0f / p;
        const float c00 = b00 * ip, c11 = b11 * ip, c22 = b22 * ip;
        const float c01 = a01 * ip, c02 = a02 * ip, c12 = a12 * ip;
        const float detB = c00 * (c11 * c22 - c12 * c12)
                         - c01 * (c01 * c22 - c12 * c02)
                         + c02 * (c01 * c12 - c11 * c02);
        float r = 0.5f * detB;
        r = fminf(1.0f, fmaxf(-1.0f, r));
        const float phi = acosf(r) * (1.0f / 3.0f);
        lmin = q3 + 2.0f * p * cosf(phi + 2.0943951023931953f); // +2pi/3 -> smallest
    } else {
        lmin = q3; // covariance ~ isotropic
    }

    // ---- eigenvector: largest cross product among rows of (A - lmin*I) ----
    const float m00 = a00 - lmin, m11 = a11 - lmin, m22 = a22 - lmin;
    float v0x = a01 * a12 - a02 * m11;            // r0 x r1
    float v0y = a02 * a01 - m00 * a12;
    float v0z = m00 * m11 - a01 * a01;
    float v1x = a01 * m22 - a02 * a12;            // r0 x r2
    float v1y = a02 * a02 - m00 * m22;
    float v1z = m00 * a12 - a01 * a02;
    float v2x = m11 * m22 - a12 * a12;            // r1 x r2
    float v2y = a12 * a02 - a01 * m22;
    float v2z = a01 * a12 - m11 * a02;

    const float n0 = v0x * v0x + v0y * v0y + v0z * v0z;
    const float n1 = v1x * v1x + v1y * v1y + v1z * v1z;
    const float n2 = v2x * v2x + v2y * v2y + v2z * v2z;

    float vx = v0x, vy = v0y, vz = v0z, nn = n0;
    if (n1 > nn) { vx = v1x; vy = v1y; vz = v1z; nn = n1; }
    if (n2 > nn) { vx = v2x; vy = v2y; vz = v2z; nn = n2; }
    if (nn > 1e-30f) {
        const float inv = rsqrtf(nn);
        vx *= inv; vy *= inv; vz *= inv;
    } else {
        vx = 1.0f; vy = 0.0f; vz = 0.0f; // fully degenerate neighborhood
    }

    if (half == 0) {
        float* o = normals + ((size_t)b * NPTS + q) * 3;
        o[0] = vx; o[1] = vy; o[2] = vz;
    }
}

// ---- deterministic two-stage loss reduction ----
__global__ __launch_bounds__(256)
void loss_partial_kernel(const float* __restrict__ nP, const float* __restrict__ nG,
                         float* __restrict__ partial, int total) {
    __shared__ float sdata[256];
    float acc = 0.0f;
    for (int i = blockIdx.x * blockDim.x + threadIdx.x; i < total;
         i += gridDim.x * blockDim.x) {
        const float ax = nP[3 * i], ay = nP[3 * i + 1], az = nP[3 * i + 2];
        const float bx = nG[3 * i], by = nG[3 * i + 1], bz = nG[3 * i + 2];
        const float dot = ax * bx + ay * by + az * bz;
        const float na  = sqrtf(ax * ax + ay * ay + az * az);
        const float nb  = sqrtf(bx * bx + by * by + bz * bz);
        const float cosv = dot / fmaxf(na * nb, EPSV);
        acc += 1.0f - fabsf(cosv);
    }
    sdata[threadIdx.x] = acc;
    __syncthreads();
#pragma unroll
    for (int s = 128; s > 0; s >>= 1) {
        if (threadIdx.x < s) sdata[threadIdx.x] += sdata[threadIdx.x + s];
        __syncthreads();
    }
    if (threadIdx.x == 0) partial[blockIdx.x] = sdata[0];
}

__global__ __launch_bounds__(64)
void loss_final_kernel(const float* __restrict__ partial, float* __restrict__ out,
                       int n, float scale) {
    __shared__ float sdata[64];
    sdata[threadIdx.x] = (threadIdx.x < n) ? partial[threadIdx.x] : 0.0f;
    __syncthreads();
#pragma unroll
    for (int s = 32; s > 0; s >>= 1) {
        if (threadIdx.x < s) sdata[threadIdx.x] += sdata[threadIdx.x + s];
        __syncthreads();
    }
    if (threadIdx.x == 0) out[0] = sdata[0] * scale;
}

extern "C" void kernel_launch(void* const* d_in, const int* in_sizes, int n_in,
                              void* d_out, int out_size, void* d_ws, size_t ws_size,
                              hipStream_t stream) {
    const float* pred = (const float*)d_in[0]; // (8, 3, 4096) f32
    const float* gt   = (const float*)d_in[1]; // (8, 3, 4096) f32

    float* ws      = (float*)d_ws;
    float* nP      = ws;                                   // (8, 4096, 3)
    float* nG      = nP + (size_t)NB * NPTS * 3;           // (8, 4096, 3)
    float* partial = nG + (size_t)NB * NPTS * 3;           // (64)

    dim3 grid(NPTS / (16 * WAVES), NB);                    // (32, 8)
    knn_normals_kernel<<<grid, BLOCKT, 0, stream>>>(pred, nP);
    knn_normals_kernel<<<grid, BLOCKT, 0, stream>>>(gt,   nG);

    const int total = NB * NPTS;
    loss_partial_kernel<<<64, 256, 0, stream>>>(nP, nG, partial, total);
    loss_final_kernel<<<1, 64, 0, stream>>>(partial, (float*)d_out, 64,
                                            1.0f / (float)total);
}